// torch_GCN_23630910062645
// MI455X (gfx1250) — compile-verified
//
#include <hip/hip_runtime.h>

// Problem constants (from reference setup_inputs)
#define N_NODES 100000
#define N_EDGES 1600000
#define D       128        // D_IN == D_OUT

typedef __attribute__((ext_vector_type(2))) float v2f;
typedef __attribute__((ext_vector_type(8))) float v8f;

#define LDS_STRIDE 132     // 128 + 4 pad: bank-conflict-free b64 reads, 16B-aligned b128 stores

// ---------------------------------------------------------------------------
// Kernel 1: h = x @ W^T + bias   using fp32 WMMA (16x16x4)
// Block = 256 threads (8 waves). Each block computes 128 rows x 128 cols.
// Each wave computes a 16x128 strip: 8 accumulator tiles of 16x16 f32.
// ---------------------------------------------------------------------------
__global__ __launch_bounds__(256) void gcn_linear_wmma(
    const float* __restrict__ x, const float* __restrict__ W,
    const float* __restrict__ bias, float* __restrict__ h)
{
    __shared__ float ldsW[D * LDS_STRIDE];

    const int tid = threadIdx.x;

    // Cooperative load of W (128x128 f32 = 64KB) into LDS, padded stride.
    // 256 threads * float4 = 4KB per iter, 16 iters.
#pragma unroll
    for (int i = 0; i < 16; ++i) {
        const int g = i * 1024 + tid * 4;
        const int r = g >> 7, c = g & 127;
        const float4 wv = *(const float4*)(W + g);
        *(float4*)(&ldsW[r * LDS_STRIDE + c]) = wv;
    }
    __syncthreads();

    const int lane  = tid & 31;
    const int wave  = tid >> 5;
    const int rBase = blockIdx.x * 128 + wave * 16;
    const int half  = lane >> 4;    // 0: lanes 0-15, 1: lanes 16-31
    const int l15   = lane & 15;

    // A-fragment row for this lane (ISA: A 16x4 f32 -> lane = M, half selects K pair)
    int arow = rBase + l15;
    if (arow >= N_NODES) arow = N_NODES - 1;   // clamp (stores are guarded)
    const float* __restrict__ xrow = x + (size_t)arow * D;

    v8f acc[8];
#pragma unroll
    for (int nt = 0; nt < 8; ++nt) acc[nt] = (v8f)0.0f;

#if __has_builtin(__builtin_amdgcn_wmma_f32_16x16x4_f32)
    for (int k0 = 0; k0 < D; k0 += 4) {
        const int kk = k0 + 2 * half;          // this lane's K pair
        const v2f a = *(const v2f*)(xrow + kk);
#pragma unroll
        for (int nt = 0; nt < 8; ++nt) {
            const int n = nt * 16 + l15;       // B-fragment: lane = N column
            const v2f bf = *(const v2f*)(&ldsW[n * LDS_STRIDE + kk]);
            acc[nt] = __builtin_amdgcn_wmma_f32_16x16x4_f32(
                false, a, false, bf, (short)0, acc[nt], false, false);
        }
    }
#else
    // VALU fallback (same accumulator layout as the WMMA C/D layout)
    for (int nt = 0; nt < 8; ++nt) {
        const int n = nt * 16 + l15;
        for (int r = 0; r < 8; ++r) {
            int row = rBase + r + half * 8;
            if (row >= N_NODES) row = N_NODES - 1;
            float s = 0.0f;
            for (int k = 0; k < D; ++k)
                s += x[(size_t)row * D + k] * ldsW[n * LDS_STRIDE + k];
            acc[nt][r] = s;
        }
    }
#endif

    // Epilogue: C/D layout -> VGPR r holds M = r (lanes 0-15) / r+8 (lanes 16-31),
    // lane (mod 16) holds N. Add bias, store.
#pragma unroll
    for (int nt = 0; nt < 8; ++nt) {
        const int col = nt * 16 + l15;
        const float bv = bias[col];
#pragma unroll
        for (int r = 0; r < 8; ++r) {
            const int row = rBase + r + half * 8;
            if (row < N_NODES)
                h[(size_t)row * D + col] = acc[nt][r] + bv;
        }
    }
}

// ---------------------------------------------------------------------------
// Kernel 2: zero the output accumulator
// ---------------------------------------------------------------------------
__global__ __launch_bounds__(256) void gcn_zero(float4* __restrict__ out, int n4)
{
    const int i = blockIdx.x * blockDim.x + threadIdx.x;
    if (i < n4) out[i] = make_float4(0.f, 0.f, 0.f, 0.f);
}

// ---------------------------------------------------------------------------
// Kernel 3: COO scatter-add. 32 lanes x float4 = one edge's 128 features.
// h[src] reads are coalesced and L2-resident (51.2MB << 192MB L2);
// atomics are coalesced f32 adds resolved at L2.
// ---------------------------------------------------------------------------
__global__ __launch_bounds__(256) void gcn_scatter(
    const float* __restrict__ h, const float* __restrict__ ew,
    const int* __restrict__ src, const int* __restrict__ dst,
    float* __restrict__ out)
{
    const long long g = (long long)blockIdx.x * blockDim.x + threadIdx.x;
    const int e = (int)(g >> 5);
    const int q = (int)(g & 31);
    if (e >= N_EDGES) return;

    const int   s = src[e];
    const int   d = dst[e];
    const float w = ew[e];

    const float4 hv = ((const float4*)h)[(size_t)s * (D / 4) + q];
    float* o = out + (size_t)d * D + q * 4;
    atomicAdd(o + 0, w * hv.x);
    atomicAdd(o + 1, w * hv.y);
    atomicAdd(o + 2, w * hv.z);
    atomicAdd(o + 3, w * hv.w);
}

// ---------------------------------------------------------------------------
// Kernel 4: ReLU in place
// ---------------------------------------------------------------------------
__global__ __launch_bounds__(256) void gcn_relu(float4* __restrict__ out, int n4)
{
    const int i = blockIdx.x * blockDim.x + threadIdx.x;
    if (i < n4) {
        float4 v = out[i];
        v.x = fmaxf(v.x, 0.f); v.y = fmaxf(v.y, 0.f);
        v.z = fmaxf(v.z, 0.f); v.w = fmaxf(v.w, 0.f);
        out[i] = v;
    }
}

// ---------------------------------------------------------------------------
extern "C" void kernel_launch(void* const* d_in, const int* in_sizes, int n_in,
                              void* d_out, int out_size, void* d_ws, size_t ws_size,
                              hipStream_t stream)
{
    const float* x  = (const float*)d_in[0];   // [N_NODES, D]
    const float* W  = (const float*)d_in[1];   // [D, D]
    const float* b  = (const float*)d_in[2];   // [D]
    const float* ew = (const float*)d_in[3];   // [N_EDGES]
    const int* esrc = (const int*)d_in[4];     // [N_EDGES]
    const int* edst = (const int*)d_in[5];     // [N_EDGES]
    float* out      = (float*)d_out;           // [N_NODES, D]
    float* h        = (float*)d_ws;            // scratch: N_NODES*D*4 = 51.2MB

    (void)in_sizes; (void)n_in; (void)out_size; (void)ws_size;

    // 1) Linear layer via fp32 WMMA
    const int gemm_blocks = (N_NODES + 127) / 128;   // 782
    gcn_linear_wmma<<<gemm_blocks, 256, 0, stream>>>(x, W, b, h);

    // 2) Zero accumulator
    const int n4 = (N_NODES * D) / 4;                // 3.2M float4
    gcn_zero<<<(n4 + 255) / 256, 256, 0, stream>>>((float4*)out, n4);

    // 3) Weighted scatter-add over edges
    const long long scatter_threads = (long long)N_EDGES * 32;
    const int scatter_blocks = (int)((scatter_threads + 255) / 256); // 200,000
    gcn_scatter<<<scatter_blocks, 256, 0, stream>>>(h, ew, esrc, edst, out);

    // 4) ReLU
    gcn_relu<<<(n4 + 255) / 256, 256, 0, stream>>>((float4*)out, n4);
}